// HarmonicSynthesizer_33088428048441
// MI455X (gfx1250) — compile-verified
//
#include <hip/hip_runtime.h>
#include <math.h>
#include <stdint.h>

// Problem constants (fixed by the reference harness)
#define NH      100
#define SRATE   16000
#define HOP     160
#define BATCH   4
#define FRAMES  500
#define TLEN    (FRAMES * HOP)          // 80000 samples per batch
#define BT      512                     // samples per block (16 wave32 waves)
#define NWAVE   (BT / 32)
#define NBLK    ((TLEN + BT - 1) / BT)  // 157 blocks per batch
#define MAXROWS 8                       // block spans <= 6 amp rows; 8 for slack

// align_corners resample ratio, rounded to f32 exactly like jnp (f32 * weak-f64 scalar -> f32)
#define RPOS ((float)(499.0 / 79999.0))

__device__ __forceinline__ float f0_upsample(const float* __restrict__ f0, int b, int t,
                                             int* lo_out, int* hi_out, float* w_out) {
    float pos = (float)t * RPOS;
    int lo = (int)floorf(pos);
    if (lo > FRAMES - 1) lo = FRAMES - 1;
    int hi = lo + 1;
    if (hi > FRAMES - 1) hi = FRAMES - 1;
    float w = pos - (float)lo;
    float a = f0[b * FRAMES + lo];
    float c = f0[b * FRAMES + hi];
    *lo_out = lo; *hi_out = hi; *w_out = w;
    return a * (1.0f - w) + c * w;
}

// Kernel 1: per-block partial sums of f0_up (f64 accumulation for phase accuracy)
__global__ void k_blocksum(const float* __restrict__ f0, double* __restrict__ sums) {
    const int blk = blockIdx.x, b = blockIdx.y;
    const int tid = threadIdx.x;
    const int t = blk * BT + tid;
    double v = 0.0;
    if (t < TLEN) {
        int lo, hi; float w;
        v = (double)f0_upsample(f0, b, t, &lo, &hi, &w);
    }
    __shared__ double sc[BT];
    sc[tid] = v;
    __syncthreads();
    for (int off = BT / 2; off > 0; off >>= 1) {
        if (tid < off) sc[tid] += sc[tid + off];
        __syncthreads();
    }
    if (tid == 0) sums[b * NBLK + blk] = sc[0];
}

// Kernel 2: tiny per-batch exclusive scan of the 157 block sums
__global__ void k_scan(const double* __restrict__ sums, double* __restrict__ prefix) {
    const int b = threadIdx.x;
    if (b < BATCH) {
        double acc = 0.0;
        for (int i = 0; i < NBLK; ++i) {
            prefix[b * NBLK + i] = acc;
            acc += sums[b * NBLK + i];
        }
    }
}

// Kernel 3: async-stage amp rows to LDS, wave32 shfl scan -> phase -> Chebyshev recurrence
__global__ void k_synth(const float* __restrict__ f0, const float* __restrict__ amps,
                        const double* __restrict__ prefix, float* __restrict__ out) {
    const int blk = blockIdx.x, b = blockIdx.y;
    const int tid = threadIdx.x;
    const int lane = tid & 31;
    const int wid = tid >> 5;
    const int t0 = blk * BT;
    const int t = t0 + tid;

    __shared__ __align__(16) float sA[MAXROWS * NH];
    __shared__ double wsum[NWAVE];

    // ---- frame-row range this block needs (contiguous in memory) ----
    const int tLast = (t0 + BT - 1 < TLEN - 1) ? (t0 + BT - 1) : (TLEN - 1);
    int loBase = (int)floorf((float)t0 * RPOS);
    if (loBase > FRAMES - 1) loBase = FRAMES - 1;
    int loLast = (int)floorf((float)tLast * RPOS);
    if (loLast > FRAMES - 1) loLast = FRAMES - 1;
    int hiMax = loLast + 1;
    if (hiMax > FRAMES - 1) hiMax = FRAMES - 1;
    const int nrows = hiMax - loBase + 1;          // <= 6
    const int nchunks = nrows * (NH / 4);          // 16B chunks per row = 25

    // ---- issue async global->LDS copies (ASYNCcnt path); overlaps with scan below ----
    {
        const float* gsrc = amps + (size_t)(b * FRAMES + loBase) * NH;  // 16B-aligned (400B rows)
        const uint32_t ldsBase = (uint32_t)(uintptr_t)&sA[0];
        const uint64_t gbase = (uint64_t)(uintptr_t)gsrc;
        if (tid < nchunks) {
            uint32_t ldsOff = ldsBase + (uint32_t)tid * 16u;
            uint32_t gOff = (uint32_t)tid * 16u;
            asm volatile("global_load_async_to_lds_b128 %0, %1, %2"
                         :: "v"(ldsOff), "v"(gOff), "s"(gbase)
                         : "memory");
        }
    }

    // ---- per-sample f0 upsample (f32, matching reference) ----
    int lo = 0, hi = 0; float w = 0.0f, f0f = 0.0f;
    double v = 0.0;
    if (t < TLEN) {
        f0f = f0_upsample(f0, b, t, &lo, &hi, &w);
        v = (double)f0f;
    }

    // ---- inclusive scan over the block: wave32 shfl scan + wave-sum scan (2 barriers) ----
    double x = v;
    #pragma unroll
    for (int off = 1; off < 32; off <<= 1) {
        double y = __shfl_up(x, (unsigned)off, 32);
        if (lane >= off) x += y;
    }
    if (lane == 31) wsum[wid] = x;
    __syncthreads();
    if (wid == 0 && lane < NWAVE) {
        double y = wsum[lane];
        #pragma unroll
        for (int off = 1; off < NWAVE; off <<= 1) {
            double z = __shfl_up(y, (unsigned)off, 32);
            if (lane >= off) y += z;
        }
        wsum[lane] = y;  // inclusive scan of wave totals
    }
    __syncthreads();
    const double incl = x + ((wid == 0) ? 0.0 : wsum[wid - 1]);

    // ---- complete async staging, publish LDS to all waves ----
    asm volatile("s_wait_asynccnt 0x0" ::: "memory");
    __syncthreads();

    if (t >= TLEN) return;  // all barriers done

    // Base phase phi = (2*pi/SR) * cumsum(f0_up); mod-2pi reduction kept in f64
    const double TWO_PI = 6.2831853071795864769252867665590;
    double total = prefix[b * NBLK + blk] + incl;
    double ph = total * (TWO_PI / (double)SRATE);
    double k = floor(ph * (1.0 / TWO_PI));
    float phr = (float)(ph - k * TWO_PI);   // [0, 2pi): safe for f32 seed

    float s1, c1;
    sincosf(phr, &s1, &c1);
    const float c2 = 2.0f * c1;

    const float* L0 = &sA[(lo - loBase) * NH];
    const float* L1 = &sA[(hi - loBase) * NH];
    const float w1 = w, w0 = 1.0f - w;

    // sum_h sin(h*phi) * lerp(A0,A1)[h] while h*f0 < Nyquist.
    // Process 4 harmonics per ds_load_b128 pair; mask per-h; monotone mask -> chunk break.
    float sm1 = 0.0f;   // sin(0*phi)
    float s   = s1;     // sin(1*phi)
    float acc = 0.0f;
    for (int hb = 1; hb <= NH; hb += 4) {
        const float4 a0 = *(const float4*)&L0[hb - 1];  // 16B aligned
        const float4 a1 = *(const float4*)&L1[hb - 1];
        const float e0[4] = {a0.x, a0.y, a0.z, a0.w};
        const float e1[4] = {a1.x, a1.y, a1.z, a1.w};
        #pragma unroll
        for (int j = 0; j < 4; ++j) {
            float freq = f0f * (float)(hb + j);
            float amp = w0 * e0[j] + w1 * e1[j];
            if (freq < 8000.0f) acc = fmaf(s, amp, acc);
            float nxt = fmaf(c2, s, -sm1);  // sin((h+1)p) = 2cos(p)sin(hp) - sin((h-1)p)
            sm1 = s;
            s = nxt;
        }
        if (f0f * (float)(hb + 3) >= 8000.0f) break;  // all later h are masked too
    }
    out[(size_t)b * TLEN + t] = acc;
}

extern "C" void kernel_launch(void* const* d_in, const int* in_sizes, int n_in,
                              void* d_out, int out_size, void* d_ws, size_t ws_size,
                              hipStream_t stream) {
    const float* f0   = (const float*)d_in[0];
    const float* amps = (const float*)d_in[1];
    float* out = (float*)d_out;

    double* sums   = (double*)d_ws;           // BATCH*NBLK doubles
    double* prefix = sums + BATCH * NBLK;     // BATCH*NBLK doubles  (~10 KB total)

    dim3 grid(NBLK, BATCH);
    k_blocksum<<<grid, BT, 0, stream>>>(f0, sums);
    k_scan<<<1, 32, 0, stream>>>(sums, prefix);
    k_synth<<<grid, BT, 0, stream>>>(f0, amps, prefix, out);
}